// AttentionGate_61503931678841
// MI455X (gfx1250) — compile-verified
//
#include <hip/hip_runtime.h>
#include <stdint.h>

typedef float v4f __attribute__((ext_vector_type(4)));
typedef int   v4i __attribute__((ext_vector_type(4)));

#define CCH   256              // channels
#define NPW   16               // pixels processed per wave (16KB per stream)
#define WPB   8                // waves per block
#define BLOCK (WPB * 32)

// ---------------- gfx1250 async global->LDS copy (ASYNCcnt path) -------------
__device__ __forceinline__ void async_g2l_b128(uint32_t lds_byte_off,
                                               const void* gaddr) {
#if __has_builtin(__builtin_amdgcn_global_load_async_to_lds_b128)
  // probed signature: (int4 addrspace(1)* src, int4 addrspace(3)* dst,
  //                    constant int offset, constant int cpol)
  typedef __attribute__((address_space(1))) v4i glb_v4i;
  typedef __attribute__((address_space(3))) v4i lds_v4i;
  __builtin_amdgcn_global_load_async_to_lds_b128(
      (glb_v4i*)(uintptr_t)gaddr,
      (lds_v4i*)(uint32_t)lds_byte_off,
      /*imm offset*/0, /*cpol*/0);
#else
  asm volatile("global_load_async_to_lds_b128 %0, %1, off"
               :: "v"(lds_byte_off), "v"((uint64_t)(uintptr_t)gaddr)
               : "memory");
#endif
}

#if __has_builtin(__builtin_amdgcn_s_wait_asynccnt)
#define WAIT_ASYNC(n) __builtin_amdgcn_s_wait_asynccnt(n)
#else
#define WAIT_ASYNC(n) asm volatile("s_wait_asynccnt " #n ::: "memory")
#endif

// -----------------------------------------------------------------------------
__global__ __launch_bounds__(BLOCK) void attn_gate_kernel(
    const float* __restrict__ x,  const float* __restrict__ g,
    const float* __restrict__ wt, const float* __restrict__ bt,
    const float* __restrict__ wg, const float* __restrict__ bg,
    const float* __restrict__ w2, const float* __restrict__ b2,
    float* __restrict__ out, int npix)
{
  // double-buffered gating staging: 2 buffers x 8 waves x 1KB = 16KB LDS
  __shared__ __attribute__((aligned(16))) float gbuf[2][WPB][CCH];

  const int lane = (int)(threadIdx.x & 31u);
  const int wid  = (int)(threadIdx.x >> 5);
  const long long wave = (long long)blockIdx.x * WPB + wid;
  const long long pix0 = wave * NPW;
  if (pix0 >= npix) return;

  const int coff = lane * 8;  // this lane owns channels [coff, coff+8)

  // per-lane weight chunks, loaded once (reused across NPW pixels)
  const v4f wt0 = *(const v4f*)(wt + coff);
  const v4f wt1 = *(const v4f*)(wt + coff + 4);
  const v4f wg0 = *(const v4f*)(wg + coff);
  const v4f wg1 = *(const v4f*)(wg + coff + 4);
  const float bias = bt[0] + bg[0];
  const float s2 = w2[0], sb = b2[0];

  // LDS byte offsets for this lane's 32-byte slot in each buffer
  const uint32_t lbase0      = (uint32_t)(uintptr_t)(&gbuf[0][wid][0])
                             + (uint32_t)(coff * 4);
  const uint32_t lbuf_stride = (uint32_t)(WPB * CCH * sizeof(float));

  // streaming pointers (advance by one pixel-row per iteration)
  const float* xp = x   + (size_t)pix0 * CCH + coff;
  const float* gp = g   + (size_t)pix0 * CCH + coff;  // points at pixel i+1's row after prologue
  float*       op = out + (size_t)pix0 * CCH + coff;
  const long long nleft = npix - pix0;                // >= 1 here

  // ---- prologue: async-stage gating row of the first pixel into buffer 0 ----
  async_g2l_b128(lbase0,      gp);
  async_g2l_b128(lbase0 + 16, gp + 4);
  gp += CCH;

#pragma unroll 4
  for (int i = 0; i < NPW; ++i) {
    if (i >= nleft) break;
    const bool have_next = (i + 1 < NPW) && (i + 1 < nleft);

    // issue next pixel's gating copy into the other buffer (pipeline depth 2)
    if (have_next) {
      const uint32_t loff = lbase0 + (uint32_t)((i + 1) & 1) * lbuf_stride;
      async_g2l_b128(loff,      gp);
      async_g2l_b128(loff + 16, gp + 4);
      __builtin_prefetch(xp + CCH, 0, 1);   // pull next input row toward L2/L0
    }
    gp += CCH;

    // stream this pixel's input channels straight into VGPRs (single touch)
    v4f x0 = __builtin_nontemporal_load((const v4f*)xp);
    v4f x1 = __builtin_nontemporal_load((const v4f*)(xp + 4));
    xp += CCH;

    // wait for the current pixel's 2 async copies (next pixel's stay in flight)
    if (have_next) { WAIT_ASYNC(2); } else { WAIT_ASYNC(0); }

    const v4f* lp = (const v4f*)(&gbuf[i & 1][wid][coff]);
    v4f g0 = lp[0];
    v4f g1 = lp[1];

    // per-lane partial of both 256-length dot products (8 channels each)
    v4f acc = x0 * wt0 + x1 * wt1 + g0 * wg0 + g1 * wg1;
    float p = (acc.x + acc.y) + (acc.z + acc.w);

    // wave32 butterfly reduction -> all lanes hold the full 256-channel sum
    p += __shfl_xor(p, 16, 32);
    p += __shfl_xor(p,  8, 32);
    p += __shfl_xor(p,  4, 32);
    p += __shfl_xor(p,  2, 32);
    p += __shfl_xor(p,  1, 32);

    // fused bias + relu + (1->1 conv) + sigmoid
    const float t  = fmaxf(p + bias, 0.0f);
    const float zz = fmaf(t, s2, sb);
    const float a  = 1.0f / (1.0f + __expf(-zz));

    // gate the input, NT store (output stream > 192MB L2; keep L2 for reads)
    __builtin_nontemporal_store(x0 * a, (v4f*)op);
    __builtin_nontemporal_store(x1 * a, (v4f*)(op + 4));
    op += CCH;
  }
}

// -----------------------------------------------------------------------------
extern "C" void kernel_launch(void* const* d_in, const int* in_sizes, int n_in,
                              void* d_out, int out_size, void* d_ws, size_t ws_size,
                              hipStream_t stream) {
  const float* x  = (const float*)d_in[0];  // input_        [B,H,W,C]
  const float* g  = (const float*)d_in[1];  // gating_signal [B,H,W,Cg]
  const float* wt = (const float*)d_in[2];  // [C]
  const float* bt = (const float*)d_in[3];  // scalar
  const float* wg = (const float*)d_in[4];  // [Cg]
  const float* bg = (const float*)d_in[5];  // scalar
  const float* w2 = (const float*)d_in[6];  // scalar
  const float* b2 = (const float*)d_in[7];  // scalar
  float* out = (float*)d_out;

  const int npix   = in_sizes[0] / CCH;                 // B*H*W = 262144
  const int waves  = (npix + NPW - 1) / NPW;
  const int blocks = (waves + WPB - 1) / WPB;

  attn_gate_kernel<<<blocks, BLOCK, 0, stream>>>(x, g, wt, bt, wg, bg, w2, b2,
                                                 out, npix);
}